// BERTLatticeEmbedding_79190607003850
// MI455X (gfx1250) — compile-verified
//
#include <hip/hip_runtime.h>

// Problem constants (from reference): B=64, S=512, H=768, T=400
#define B_ 64
#define S_ 512
#define H_ 768
#define T_ 400

#define NCHUNK 8
#define CHUNK (S_ / NCHUNK)   // 64 s-steps per block
#define TPB 192               // threads per block = H_/4 (one float4 per thread)
#define GRAN 8                // rows per async granule
#define NGRAN (CHUNK / GRAN)  // 8 granules per chunk

#if defined(__gfx1250__) && \
    __has_builtin(__builtin_amdgcn_global_load_async_to_lds_b128) && \
    __has_builtin(__builtin_amdgcn_s_wait_asynccnt)
#define USE_ASYNC 1
#else
#define USE_ASYNC 0
#endif

typedef int v4i __attribute__((__vector_size__(16)));

#if USE_ASYNC
#define AS1 __attribute__((address_space(1)))
#define AS3 __attribute__((address_space(3)))
#endif

// Ownership scheme (word_ids sorted per sample => each word is a contiguous
// run in S):
//  * A chunk owns every run that STARTS inside [s0, s1). If the run crosses
//    s1, the owner keeps streaming rows past s1 straight from global and
//    stores the complete mean; later chunks skip that leading run entirely.
//  * Gap words (zero pieces) are zero-stored by the chunk owning the
//    transition: internal transitions, plus each chunk's right-edge
//    transition; chunk 0 zeroes [0, w[0]), the last chunk zeroes the tail.
// => every output element written exactly once; no atomics; no pre-zero pass.
__global__ __launch_bounds__(TPB) void pool_kernel(const float* __restrict__ hidden,
                                                   const int* __restrict__ word_ids,
                                                   float* __restrict__ out) {
  const int b   = blockIdx.x;      // sample
  const int c   = blockIdx.y;      // S-chunk
  const int s0  = c * CHUNK;
  const int s1  = s0 + CHUNK;
  const int tid = threadIdx.x;     // owns float4 column tid (h = tid*4)

  const float4* __restrict__ hrow = (const float4*)hidden + (size_t)b * S_ * TPB + tid;
  const int*    __restrict__ w    = word_ids + b * S_;
  float*        __restrict__ outb = out + (size_t)b * T_ * H_ + tid * 4;

  float ax = 0.f, ay = 0.f, az = 0.f, aw = 0.f;
  int  cur       = w[s0];
  int  run_start = s0;
  bool skipping  = (s0 > 0) && (w[s0 - 1] == cur);  // leading run owned by left chunk

  // Zero-store all gap words strictly between lo and hi.
  auto zero_gap = [&](int lo, int hi) {
    float4 z; z.x = 0.f; z.y = 0.f; z.z = 0.f; z.w = 0.f;
    for (int t = lo + 1; t < hi; ++t) *(float4*)(outb + (size_t)t * H_) = z;
  };

  // Store mean of owned run [run_start, re); if re==s1 the run may continue
  // past the chunk edge -- stream those rows directly from global.
  auto flush = [&](int word, int re) {
    int count = re - run_start;
    if (re == s1) {
      int s = s1;
      while (s < S_ && w[s] == word) {
        float4 v = hrow[(size_t)s * TPB];
        ax += v.x; ay += v.y; az += v.z; aw += v.w;
        ++count; ++s;
      }
    }
    const float fc = (float)count;
    float4 r; r.x = ax / fc; r.y = ay / fc; r.z = az / fc; r.w = aw / fc;
    *(float4*)(outb + (size_t)word * H_) = r;
    ax = 0.f; ay = 0.f; az = 0.f; aw = 0.f;
  };

#if USE_ASYNC
  // Double-buffered async global->LDS staging. Each thread reads back only
  // the LDS bytes its own wave's async loads wrote (wave-private columns), so
  // ASYNCcnt alone sequences the pipeline -- no workgroup barrier needed.
  __shared__ float4 buf[2 * GRAN * TPB];  // 48 KB
  AS3 char* lds_base = (AS3 char*)buf;

  auto issue = [&](int g) {
    const float4* src = hrow + (size_t)(s0 + g * GRAN) * TPB;
    AS3 char* dst = lds_base + (size_t)(((g & 1) * GRAN * TPB) + tid) * 16;
#pragma unroll
    for (int i = 0; i < GRAN; ++i) {
      __builtin_amdgcn_global_load_async_to_lds_b128(
          (AS1 v4i*)(src + (size_t)i * TPB),
          (AS3 v4i*)(dst + (size_t)i * TPB * 16),
          0, 0);
    }
  };

  issue(0);                         // first granule in flight ASAP...
  if (s0 == 0) zero_gap(-1, cur);   // ...overlapped with head-gap stores

  for (int g = 0; g < NGRAN; ++g) {
    if (g + 1 < NGRAN) {
      issue(g + 1);
      __builtin_amdgcn_s_wait_asynccnt(GRAN);  // granule g landed in LDS
    } else {
      __builtin_amdgcn_s_wait_asynccnt(0);
    }
    const float4* lbuf = buf + (g & 1) * GRAN * TPB + tid;
#pragma unroll
    for (int i = 0; i < GRAN; ++i) {
      const int s  = s0 + g * GRAN + i;
      const int ws = w[s];                      // wave-uniform -> scalar load
      if (ws != cur) {
        if (skipping) skipping = false;
        else flush(cur, s);
        zero_gap(cur, ws);
        cur = ws; run_start = s;
      }
      float4 v = lbuf[i * TPB];                 // ds_load_b128, conflict-free
      if (!skipping) { ax += v.x; ay += v.y; az += v.z; aw += v.w; }
    }
  }
#else
  // Fallback: direct coalesced b128 streaming with explicit prefetch
  // (global_prefetch_b8 on gfx1250).
  if (s0 == 0) zero_gap(-1, cur);
  for (int s = s0; s < s1; ++s) {
    if (s + 16 < S_) __builtin_prefetch(hrow + (size_t)(s + 16) * TPB, 0, 0);
    const int ws = w[s];
    if (ws != cur) {
      if (skipping) skipping = false;
      else flush(cur, s);
      zero_gap(cur, ws);
      cur = ws; run_start = s;
    }
    float4 v = hrow[(size_t)s * TPB];
    if (!skipping) { ax += v.x; ay += v.y; az += v.z; aw += v.w; }
  }
#endif

  // Epilogue: flush the last owned run (with right extension), then handle
  // this chunk's right-edge transition / tail gap.
  if (!skipping) flush(cur, s1);
  const int next = (s1 < S_) ? w[s1] : T_;
  if (next != cur) zero_gap(cur, next);
}

extern "C" void kernel_launch(void* const* d_in, const int* in_sizes, int n_in,
                              void* d_out, int out_size, void* d_ws, size_t ws_size,
                              hipStream_t stream) {
  const float* hidden   = (const float*)d_in[0];
  const int*   word_ids = (const int*)d_in[1];
  // d_in[2] = num_tokens scalar (400) -- baked into T_.
  float* out = (float*)d_out;

  dim3 grid(B_, NCHUNK);  // 512 blocks x 192 threads
  pool_kernel<<<grid, TPB, 0, stream>>>(hidden, word_ids, out);
}